// Denoiser_4260607558094
// MI455X (gfx1250) — compile-verified
//
#include <hip/hip_runtime.h>
#include <stdint.h>

// Denoiser: per-pixel 9x9 kernel-predicting filter.
// out[h,w,c] = sum_{ki,kj} cnn[h,w,ki*9+kj] * unet_reflectpad[h+ki, w+kj, c]
//
// Roofline: 0.51 GFLOP vs ~365MB moved -> AI ~1.4 FLOP/B, HBM-bound (~16us at
// 23.3 TB/s). Strategy: stream cnn_out (340MB, zero reuse) non-temporally as
// b128 loads; stage the 81x-reused unet tile into LDS with the CDNA5 Tensor
// Data Mover (async, TENSORcnt); 2 pixels/thread for vertical tap-row reuse;
// packed v_pk_fma_f32 for the channel xy pair.

static constexpr int IMG_H = 1024;
static constexpr int IMG_W = 1024;
static constexpr int KK    = 9;     // kernel size
static constexpr int RAD   = 4;     // KK/2
static constexpr int TILE_W = 32;   // pixels per tile, x
static constexpr int TILE_H = 16;   // pixels per tile, y (2 rows per thread)
static constexpr int PADW  = TILE_W + 2 * RAD;   // 40
static constexpr int PADH  = TILE_H + 2 * RAD;   // 24
static constexpr int SMEM_N = PADH * PADW * 3;   // 2880 floats = 11.5 KB

typedef float v4f __attribute__((ext_vector_type(4), aligned(4)));  // 4B-aligned b128
typedef float v2f __attribute__((ext_vector_type(2)));
typedef unsigned int u32x4 __attribute__((ext_vector_type(4)));
typedef int i32x8 __attribute__((ext_vector_type(8)));
typedef int i32x4 __attribute__((ext_vector_type(4)));

#if defined(__has_builtin)
#if __has_builtin(__builtin_amdgcn_tensor_load_to_lds) && \
    __has_builtin(__builtin_amdgcn_s_wait_tensorcnt)
#define USE_TDM 1
#endif
#endif

__device__ __forceinline__ int reflect_idx(int i, int n) {
  i = (i < 0) ? -i : i;                   // jnp reflect: -1 -> 1
  return (i >= n) ? (2 * n - 2 - i) : i;  // n -> n-2
}

__global__ __launch_bounds__(256)
void denoiser9x9(const float* __restrict__ unet,
                 const float* __restrict__ cnn,
                 float* __restrict__ out) {
  __shared__ float sm[SMEM_N];
  const int tid = threadIdx.x;
  const int w0 = blockIdx.x * TILE_W;
  const int h0 = blockIdx.y * TILE_H;

  const bool interior =
      (h0 >= RAD) && (h0 + TILE_H + RAD <= IMG_H) &&
      (w0 >= RAD) && (w0 + TILE_W + RAD <= IMG_W);
  (void)interior;

#ifdef USE_TDM
  if (interior) {
    // Interior tile: padded source rectangle is fully in-bounds -> one TDM
    // 2D tile DMA: 24 rows x 120 floats, row stride 3072 floats.
    if (tid < 32) {  // wave 0 issues the descriptor once
      unsigned lds_off =
          (unsigned)(uintptr_t)(__attribute__((address_space(3))) float*)sm;
      uint64_t ga = (uint64_t)(uintptr_t)(
          unet + ((uint64_t)(h0 - RAD) * IMG_W + (uint64_t)(w0 - RAD)) * 3u);
      u32x4 g0;
      g0[0] = 1u;                                   // count=1, user descriptor
      g0[1] = lds_off;                              // LDS byte address
      g0[2] = (unsigned)(ga & 0xffffffffu);         // global addr [31:0]
      g0[3] = (unsigned)((ga >> 32) & 0x01ffffffu)  // global addr [56:32]
              | (2u << 30);                         // type = 2 ("image")
      i32x8 g1;
      g1[0] = (int)(2u << 16);                      // data_size = 4 bytes
      g1[1] = (int)((unsigned)(PADW * 3) << 16);    // tensor_dim0 lo16 (=120)
      g1[2] = (int)((unsigned)PADH << 16);          // td0 hi=0 | tensor_dim1 lo16 (=24)
      g1[3] = (int)((unsigned)(PADW * 3) << 16);    // td1 hi=0 | tile_dim0 (=120)
      g1[4] = PADH;                                 // tile_dim1=24 | tile_dim2=0
      g1[5] = IMG_W * 3;                            // tensor_dim0_stride lo32 (=3072)
      g1[6] = 0;                                    // stride hi | td1_stride lo
      g1[7] = 0;
      i32x4 gz4 = {0, 0, 0, 0};                     // 2D tensor: groups 2/3 unused
      i32x8 gz8 = {0, 0, 0, 0, 0, 0, 0, 0};
      __builtin_amdgcn_tensor_load_to_lds(g0, g1, gz4, gz4, gz8, 0);
      __builtin_amdgcn_s_wait_tensorcnt(0);
    }
  } else
#endif
  {
    // Border tiles (and non-TDM builds): cooperative stage with reflect.
    for (int e = tid; e < SMEM_N; e += 256) {
      int c = e % 3;
      int t = e / 3;
      int col = t % PADW;
      int row = t / PADW;
      int gr = reflect_idx(h0 - RAD + row, IMG_H);
      int gc = reflect_idx(w0 - RAD + col, IMG_W);
      sm[e] = unet[((uint64_t)gr * IMG_W + (uint64_t)gc) * 3u + (unsigned)c];
    }
  }
  __syncthreads();

  // Each thread computes a vertical pair of pixels (rows share 8/9 tap rows).
  const int tx = tid & 31;            // pixel col within tile
  const int tg = tid >> 5;            // 0..7
  const int px = w0 + tx;
  const int py = h0 + 2 * tg;

  const float* wA = cnn + ((uint64_t)py * IMG_W + (uint64_t)px) * (uint64_t)(KK * KK);
  const float* wB = wA + (uint64_t)IMG_W * (uint64_t)(KK * KK);  // pixel one row below

  v2f a0xy = {0.f, 0.f};  float a0z = 0.f;   // pixel0 accumulators (ch0,1 packed)
  v2f a1xy = {0.f, 0.f};  float a1z = 0.f;   // pixel1 accumulators
  const int lr0 = 2 * tg;             // padded-tile row of pixel0's first tap row

#pragma unroll
  for (int pr = 0; pr < KK + 1; ++pr) {   // 10 shared patch rows
    const float* srow = &sm[((lr0 + pr) * PADW + tx) * 3];
    float u[3 * KK];
#pragma unroll
    for (int j = 0; j < 3 * KK; ++j) u[j] = srow[j];

    if (pr < KK) {                    // pixel0 tap row ki = pr
      const float* wp = wA + pr * KK;
      v4f wlo = __builtin_nontemporal_load((const v4f*)wp);
      v4f whi = __builtin_nontemporal_load((const v4f*)(wp + 4));
      float w8 = __builtin_nontemporal_load(wp + 8);
      float wv[KK] = {wlo.x, wlo.y, wlo.z, wlo.w,
                      whi.x, whi.y, whi.z, whi.w, w8};
#pragma unroll
      for (int j = 0; j < KK; ++j) {
        v2f uxy = {u[3 * j + 0], u[3 * j + 1]};
        v2f wj  = {wv[j], wv[j]};
        a0xy = __builtin_elementwise_fma(wj, uxy, a0xy);   // v_pk_fma_f32
        a0z  = fmaf(wv[j], u[3 * j + 2], a0z);
      }
    }
    if (pr >= 1) {                    // pixel1 tap row ki = pr-1
      const float* wp = wB + (pr - 1) * KK;
      v4f wlo = __builtin_nontemporal_load((const v4f*)wp);
      v4f whi = __builtin_nontemporal_load((const v4f*)(wp + 4));
      float w8 = __builtin_nontemporal_load(wp + 8);
      float wv[KK] = {wlo.x, wlo.y, wlo.z, wlo.w,
                      whi.x, whi.y, whi.z, whi.w, w8};
#pragma unroll
      for (int j = 0; j < KK; ++j) {
        v2f uxy = {u[3 * j + 0], u[3 * j + 1]};
        v2f wj  = {wv[j], wv[j]};
        a1xy = __builtin_elementwise_fma(wj, uxy, a1xy);   // v_pk_fma_f32
        a1z  = fmaf(wv[j], u[3 * j + 2], a1z);
      }
    }
  }

  float* o0 = out + ((uint64_t)py * IMG_W + (uint64_t)px) * 3u;
  __builtin_nontemporal_store(a0xy.x, o0 + 0);
  __builtin_nontemporal_store(a0xy.y, o0 + 1);
  __builtin_nontemporal_store(a0z,    o0 + 2);
  float* o1 = o0 + (uint64_t)IMG_W * 3u;
  __builtin_nontemporal_store(a1xy.x, o1 + 0);
  __builtin_nontemporal_store(a1xy.y, o1 + 1);
  __builtin_nontemporal_store(a1z,    o1 + 2);
}

extern "C" void kernel_launch(void* const* d_in, const int* in_sizes, int n_in,
                              void* d_out, int out_size, void* d_ws, size_t ws_size,
                              hipStream_t stream) {
  (void)in_sizes; (void)n_in; (void)out_size; (void)d_ws; (void)ws_size;
  const float* unet = (const float*)d_in[0];   // [1024,1024,3] f32
  const float* cnn  = (const float*)d_in[1];   // [1024,1024,81] f32
  float* out = (float*)d_out;                  // [1024,1024,3] f32
  dim3 grid(IMG_W / TILE_W, IMG_H / TILE_H);   // 32 x 64 tiles
  denoiser9x9<<<grid, 256, 0, stream>>>(unet, cnn, out);
}